// RGNN_15470472200582
// MI455X (gfx1250) — compile-verified
//
#include <hip/hip_runtime.h>
#include <hip/hip_bf16.h>

// ---------------- problem constants (match reference) ----------------
#define BQ    1024
#define DIMQ  64
#define LQ    2
#define NRELQ 4
#define NQ    131072
#define EQ    524288
#define DFMQ  384

typedef float v2f __attribute__((ext_vector_type(2)));
typedef float v8f __attribute__((ext_vector_type(8)));

// orderable-uint encoding for float atomic max
__device__ __forceinline__ unsigned fenc(float f) {
    unsigned b = __float_as_uint(f);
    return b ^ ((((int)b) >> 31) | 0x80000000u);
}
__device__ __forceinline__ float fdec(unsigned u) {
    unsigned b = u ^ (((int)u >= 0) ? 0xFFFFFFFFu : 0x80000000u);
    return __uint_as_float(b);
}

// ---------------------------------------------------------------------
// Generic 64-K GEMM: out[r, ooff + 0..63] = act(X[r,0:64] @ W + bias)
// Block = 256 threads (8 waves), 128 rows/block, WMMA f32 16x16x4.
// W is staged in LDS in K-pair interleaved layout so each B fragment is
// a single aligned ds_load_b64:  sW[(k>>1)*128 + col*2 + (k&1)].
// nrows must be a multiple of 128 (true for B=1024 and N=131072).
// ---------------------------------------------------------------------
__global__ __launch_bounds__(256) void mm64_wmma_k(
    const float* __restrict__ X, const float* __restrict__ W,
    const float* __restrict__ bias, float* __restrict__ out,
    int ostride, int ooff, int dorelu)
{
    __shared__ float sW[64 * 64];    // 16 KB, K-pair interleaved
    __shared__ float sX[128 * 64];   // 32 KB, row major
    const int  t  = threadIdx.x;
    const long r0 = (long)blockIdx.x * 128;

    // stage W: coalesced float4 global loads, scatter into pair layout
    const float4* W4 = (const float4*)W;
#pragma unroll
    for (int k = 0; k < 4; ++k) {
        const float4 v = W4[t + 256 * k];
        const int flat = (t + 256 * k) * 4;    // linear float index
        const int row  = flat >> 6;            // K index
        const int col  = flat & 63;            // N index (multiple of 4)
        float* dstp = &sW[(row >> 1) * 128 + col * 2 + (row & 1)];
        dstp[0] = v.x; dstp[2] = v.y; dstp[4] = v.z; dstp[6] = v.w;
    }
    // stage X tile: coalesced float4
    const float4* X4  = (const float4*)(X + r0 * 64);
    float4*       sX4 = (float4*)sX;
#pragma unroll
    for (int k = 0; k < 8; ++k) sX4[t + 256 * k] = X4[t + 256 * k];
    __syncthreads();

    const int lane = t & 31;
    const int l16  = lane & 15;
    const int hi   = lane >> 4;          // 0: K pair (k,k+1), 1: (k+2,k+3)
    const int M0   = (t >> 5) * 16;      // 16-row tile per wave

    v8f acc[4];
#pragma unroll
    for (int j = 0; j < 4; ++j)
#pragma unroll
        for (int e = 0; e < 8; ++e) acc[j][e] = 0.0f;

    const float* aRow = &sX[(M0 + l16) * 64];
#pragma unroll
    for (int k = 0; k < 64; k += 4) {
        // A fragment 16x4: lane holds (X[m][klo], X[m][klo+1]), klo = k + hi*2
        v2f a = *(const v2f*)(aRow + k + hi * 2);
        const float* bRow = &sW[((k >> 1) + hi) * 128];
#pragma unroll
        for (int j = 0; j < 4; ++j) {
            // B fragment: contiguous aligned pair {W[klo][col], W[klo+1][col]}
            v2f b = *(const v2f*)(bRow + (j * 16 + l16) * 2);
            acc[j] = __builtin_amdgcn_wmma_f32_16x16x4_f32(
                false, a, false, b, (short)0, acc[j], false, false);
        }
    }

    // C/D layout: VGPR r -> lanes0-15: M=r, lanes16-31: M=r+8; col = N tile + l16
#pragma unroll
    for (int j = 0; j < 4; ++j) {
        const int col = j * 16 + l16;
        const float bv = bias ? bias[col] : 0.0f;
#pragma unroll
        for (int r = 0; r < 8; ++r) {
            const long row = r0 + M0 + r + hi * 8;
            float v = acc[j][r] + bv;
            if (dorelu) v = fmaxf(v, 0.0f);
            out[row * (long)ostride + ooff + col] = v;
        }
    }
}

// ------------------- row gather: out[r,:] = emb[idx[r],:] -------------------
__global__ void gather_rows_k(float* __restrict__ out, const float* __restrict__ emb,
                              const int* __restrict__ idx, int nrows)
{
    const int g   = blockIdx.x * 256 + threadIdx.x;
    const int row = g >> 4, c = g & 15;
    if (row < nrows) {
        const int s = idx[row];
        ((float4*)out)[(long)row * 16 + c] = ((const float4*)emb)[(long)s * 16 + c];
    }
}

// ------------------- per-node attention dots a0 = h.att0, a1 = h.att1 -------
__global__ void node_att_k(const float* __restrict__ h, const float* __restrict__ att,
                           float* __restrict__ a0, float* __restrict__ a1, int n)
{
    __shared__ float s0[64], s1[64];
    const int t = threadIdx.x;
    if (t < 64) s0[t] = att[t];
    else if (t < 128) s1[t - 64] = att[t];
    __syncthreads();
    const int node = blockIdx.x * 256 + t;
    if (node < n) {
        const float* hr = h + (long)node * 64;
        float x0 = 0.f, x1 = 0.f;
#pragma unroll
        for (int c = 0; c < 64; ++c) { const float v = hr[c]; x0 += v * s0[c]; x1 += v * s1[c]; }
        a0[node] = x0; a1[node] = x1;
    }
}

// ------------------- rc[t] = rel[t] . att2 (t < NREL) -----------------------
__global__ void rel_att_k(const float* __restrict__ rel, const float* __restrict__ att2,
                          float* __restrict__ rc)
{
    const int t = threadIdx.x;
    if (t < NRELQ) {
        float s = 0.f;
        for (int c = 0; c < 64; ++c) s += rel[t * 64 + c] * att2[c];
        rc[t] = s;
    }
}

// ------------------- init per-node softmax state ----------------------------
__global__ void init_seg_k(unsigned* __restrict__ emax, float* __restrict__ esum, int n)
{
    const int i = blockIdx.x * 256 + threadIdx.x;
    if (i < n) { emax[i] = 0x007FFFFFu; esum[i] = 0.0f; }   // enc(-inf), 0
}

// ------------------- edge pass 1: logits + segment max ----------------------
__global__ void edge_logits_k(const int* __restrict__ src, const int* __restrict__ dst,
                              const int* __restrict__ et, const float* __restrict__ a0,
                              const float* __restrict__ a1, const float* __restrict__ rc,
                              float* __restrict__ eE, unsigned* __restrict__ emax, int ne)
{
    const int e = blockIdx.x * 256 + threadIdx.x;
    if (e < ne) {
        const int s = src[e], d = dst[e], tp = et[e];
        float v = a0[s] + a1[d] + rc[tp];
        v = v > 0.f ? v : 0.2f * v;                 // leaky_relu 0.2
        eE[e] = v;
        atomicMax(&emax[d], fenc(v));
    }
}

// ------------------- edge pass 2: exp + segment sum -------------------------
__global__ void edge_exp_k(const int* __restrict__ dst, const unsigned* __restrict__ emax,
                           float* __restrict__ eE, float* __restrict__ esum, int ne)
{
    const int e = blockIdx.x * 256 + threadIdx.x;
    if (e < ne) {
        const int d  = dst[e];
        const float ex = __expf(eE[e] - fdec(emax[d]));
        eE[e] = ex;
        atomicAdd(&esum[d], ex);
    }
}

// ------------------- edge pass 3: alpha * h[src] -> out[dst] ----------------
__global__ void edge_agg_k(const int* __restrict__ src, const int* __restrict__ dst,
                           const float* __restrict__ eE, const float* __restrict__ esum,
                           const float* __restrict__ h, float* __restrict__ out, int ne)
{
    const int g = blockIdx.x * 256 + threadIdx.x;
    const int e = g >> 4, lane = g & 15;
    if (e < ne) {
        const int d = dst[e], s = src[e];
        const float alpha = eE[e] / (esum[d] + 1e-16f);
        const float4 hv = ((const float4*)h)[(long)s * 16 + lane];
        float* op = out + (long)d * 64 + lane * 4;
        atomicAdd(op + 0, alpha * hv.x);
        atomicAdd(op + 1, alpha * hv.y);
        atomicAdd(op + 2, alpha * hv.z);
        atomicAdd(op + 3, alpha * hv.w);
    }
}

__global__ void relu_k(float* __restrict__ x, int n4)
{
    const int i = blockIdx.x * 256 + threadIdx.x;
    if (i < n4) {
        float4 v = ((float4*)x)[i];
        v.x = fmaxf(v.x, 0.f); v.y = fmaxf(v.y, 0.f);
        v.z = fmaxf(v.z, 0.f); v.w = fmaxf(v.w, 0.f);
        ((float4*)x)[i] = v;
    }
}

// ------------------- gated pooling: one block per graph ---------------------
// score = sigmoid(dot(p[n], q[b]) / 8); x[n] *= score; pooled[b] = colmax(x)
__global__ __launch_bounds__(128) void pool_k(const float* __restrict__ p,
                                              const float* __restrict__ q,
                                              float* __restrict__ x,
                                              float* __restrict__ pooled)
{
    __shared__ float sq[64];
    __shared__ float sv[128 * 65];   // +1 pad: avoid 32-way bank conflicts
    const int b = blockIdx.x, t = threadIdx.x;
    if (t < 64) sq[t] = q[b * 64 + t];
    __syncthreads();
    const long n  = (long)b * 128 + t;
    const float* pr = p + n * 64;
    float s = 0.f;
#pragma unroll
    for (int c = 0; c < 64; ++c) s += pr[c] * sq[c];
    const float sc = 1.0f / (1.0f + __expf(-0.125f * s));
    float* xr = x + n * 64;
#pragma unroll
    for (int c = 0; c < 64; ++c) {
        const float v = xr[c] * sc;
        xr[c] = v;                    // x scaled in place (next layer input)
        sv[t * 65 + c] = v;
    }
    __syncthreads();
    if (t < 64) {
        float m = -3.402823466e38f;
        for (int nn = 0; nn < 128; ++nn) m = fmaxf(m, sv[nn * 65 + t]);
        pooled[b * 64 + t] = m;
    }
}

// ------------------- rssq[i] = sum_j V[i,j]^2 -------------------------------
__global__ void rssq_k(const float* __restrict__ V, float* __restrict__ r)
{
    const int i = blockIdx.x * 128 + threadIdx.x;
    if (i < DFMQ) {
        float s = 0.f;
        for (int j = 0; j < DFMQ; ++j) { const float v = V[i * DFMQ + j]; s += v * v; }
        r[i] = s;
    }
}

// ------------------- FM head: one block per interaction ---------------------
__global__ __launch_bounds__(256) void fm_k(
    const float* __restrict__ xfm, const float* __restrict__ V,
    const float* __restrict__ w, const float* __restrict__ rssq,
    const int* __restrict__ uid, const int* __restrict__ iid,
    const float* __restrict__ bu, const float* __restrict__ bi,
    const float* __restrict__ b0, float* __restrict__ out)
{
    __shared__ float sx[DFMQ];
    __shared__ float sred[256];
    const int b = blockIdx.x, t = threadIdx.x;
    for (int i = t; i < DFMQ; i += 256) sx[i] = xfm[(long)b * DFMQ + i];
    __syncthreads();
    float part = 0.f;
    for (int j = t; j < DFMQ; j += 256) {
        float xv = 0.f;
        for (int i = 0; i < DFMQ; ++i) xv += sx[i] * V[i * DFMQ + j];
        part += 0.5f * xv * xv;
    }
    for (int i = t; i < DFMQ; i += 256) {
        const float xi = sx[i];
        part += xi * w[i] - 0.5f * xi * xi * rssq[i];
    }
    sred[t] = part;
    __syncthreads();
    for (int st = 128; st > 0; st >>= 1) {
        if (t < st) sred[t] += sred[t + st];
        __syncthreads();
    }
    if (t == 0) out[b] = sred[0] + bu[uid[b]] + bi[iid[b]] + b0[0];
}

// ---------------------------------------------------------------------------
extern "C" void kernel_launch(void* const* d_in, const int* in_sizes, int n_in,
                              void* d_out, int out_size, void* d_ws, size_t ws_size,
                              hipStream_t stream)
{
    (void)in_sizes; (void)n_in; (void)out_size; (void)ws_size;

    const int*   uid      = (const int*)d_in[0];
    const int*   iid      = (const int*)d_in[1];
    const float* user_emb = (const float*)d_in[10];
    const float* item_emb = (const float*)d_in[11];
    const float* word_emb = (const float*)d_in[12];
    const float* trans_w_W = (const float*)d_in[17];
    const float* trans_w_b = (const float*)d_in[18];
    const float* inter_u_W = (const float*)d_in[19];
    const float* inter_u_b = (const float*)d_in[20];
    const float* inter_i_W = (const float*)d_in[21];
    const float* inter_i_b = (const float*)d_in[22];
    const float* pool_W    = (const float*)d_in[31];
    const float* fm_w      = (const float*)d_in[32];
    const float* fm_V      = (const float*)d_in[33];
    const float* fm_bu     = (const float*)d_in[34];
    const float* fm_bi     = (const float*)d_in[35];
    const float* fm_b0     = (const float*)d_in[36];

    // ---- workspace carving (all offsets multiples of 4 floats) ----
    float* ws = (float*)d_ws;
    size_t off = 0;
    auto alloc = [&](size_t nf) { float* p = ws + off; off += nf; return p; };
    float*    x     = alloc((size_t)NQ * 64);
    float*    h     = alloc((size_t)NQ * 64);
    float*    a0    = alloc(NQ);
    float*    a1    = alloc(NQ);
    unsigned* emax  = (unsigned*)alloc(NQ);
    float*    esum  = alloc(NQ);
    float*    eE    = alloc(EQ);
    float*    rc    = alloc(4);
    float*    ue    = alloc((size_t)BQ * 64);
    float*    ie    = alloc((size_t)BQ * 64);
    float*    qem   = alloc((size_t)LQ * BQ * 64);
    float*    pooled= alloc((size_t)BQ * 64);
    float*    xfm   = alloc((size_t)BQ * DFMQ);
    float*    rssq  = alloc(DFMQ);

    auto mm = [&](const float* X, const float* W, const float* bias, float* out_,
                  int nrows, int ostride, int ooff, int relu) {
        mm64_wmma_k<<<nrows / 128, 256, 0, stream>>>(X, W, bias, out_, ostride, ooff, relu);
    };

    // ---- entity embeddings, inter reps, FM precompute ----
    gather_rows_k<<<BQ * 16 / 256, 256, 0, stream>>>(ue, user_emb, uid, BQ);
    gather_rows_k<<<BQ * 16 / 256, 256, 0, stream>>>(ie, item_emb, iid, BQ);
    mm(ue, inter_u_W, inter_u_b, xfm, BQ, DFMQ, 0,   1);   // user_rep -> cols [0,64)
    mm(ie, inter_i_W, inter_i_b, xfm, BQ, DFMQ, 192, 1);   // item_rep -> cols [192,256)
    rssq_k<<<3, 128, 0, stream>>>(fm_V, rssq);

    // ---- per side: u (side 0), i (side 1) ----
    for (int side = 0; side < 2; ++side) {
        const int*   nodes   = (const int*)d_in[side ? 3 : 2];
        const int*   ei      = (const int*)d_in[side ? 5 : 4];
        const int*   et      = (const int*)d_in[side ? 7 : 6];
        const float* entW    = (const float*)d_in[side ? 15 : 13];
        const float* entB    = (const float*)d_in[side ? 16 : 14];
        const float* convW   = (const float*)d_in[side ? 27 : 23];
        const float* convB   = (const float*)d_in[side ? 28 : 24];
        const float* convAtt = (const float*)d_in[side ? 29 : 25];
        const float* convRel = (const float*)d_in[side ? 30 : 26];
        const float* ent     = side ? ie : ue;
        const int    fmbase  = side ? 256 : 64;
        const int*   src     = ei;
        const int*   dst     = ei + EQ;

        gather_rows_k<<<NQ * 16 / 256, 256, 0, stream>>>(x, word_emb, nodes, NQ);
        for (int l = 0; l < LQ; ++l)        // query embeddings q_l = relu(ent @ trans + b)
            mm(ent, entW + l * 64 * 64, entB + l * 64, qem + (size_t)l * BQ * 64, BQ, 64, 0, 1);

        for (int l = 0; l < LQ; ++l) {
            // h = x @ convW[l] + convb[l]
            mm(x, convW + l * 64 * 64, convB + l * 64, h, NQ, 64, 0, 0);
            // attention precomputes
            rel_att_k<<<1, 32, 0, stream>>>(convRel + l * NRELQ * 64,
                                            convAtt + l * 3 * 64 + 128, rc);
            node_att_k<<<NQ / 256, 256, 0, stream>>>(h, convAtt + l * 3 * 64, a0, a1, NQ);
            // segment softmax over dst
            init_seg_k<<<NQ / 256, 256, 0, stream>>>(emax, esum, NQ);
            edge_logits_k<<<EQ / 256, 256, 0, stream>>>(src, dst, et, a0, a1, rc, eE, emax, EQ);
            edge_exp_k<<<EQ / 256, 256, 0, stream>>>(dst, emax, eE, esum, EQ);
            // aggregate alpha * h[src] into x (zeroed), then relu
            hipMemsetAsync(x, 0, (size_t)NQ * 64 * 4, stream);
            edge_agg_k<<<EQ * 16 / 256, 256, 0, stream>>>(src, dst, eE, esum, h, x, EQ);
            relu_k<<<NQ * 64 / 4 / 256, 256, 0, stream>>>(x, NQ * 64 / 4);
            // pooling: p = x @ pool_W (into h), gate by q, colmax per graph
            mm(x, pool_W, nullptr, h, NQ, 64, 0, 0);
            pool_k<<<BQ, 128, 0, stream>>>(h, qem + (size_t)l * BQ * 64, x, pooled);
            // trans_w -> FM feature slots
            mm(pooled, trans_w_W + l * 64 * 64, trans_w_b + l * 64, xfm,
               BQ, DFMQ, fmbase + l * 64, 1);
        }
    }

    // ---- FM head ----
    fm_k<<<BQ, 256, 0, stream>>>(xfm, fm_V, fm_w, rssq, uid, iid,
                                 fm_bu, fm_bi, fm_b0, (float*)d_out);
}